// GNN_69904887710173
// MI455X (gfx1250) — compile-verified
//
#include <hip/hip_runtime.h>
#include <hip/hip_bf16.h>

#define NN 100000
#define NE 3200000
#define NG 128

typedef __attribute__((ext_vector_type(2))) float v2f;
typedef __attribute__((ext_vector_type(8))) float v8f;

// ---------------- degree / norm prep ----------------

__global__ void k_deg_init(float* __restrict__ deg) {
  int i = blockIdx.x * blockDim.x + threadIdx.x;
  if (i < NN) deg[i] = 1.0f;  // self loop
}

__global__ void k_deg_edges(const int* __restrict__ dst, float* __restrict__ deg) {
  int e = blockIdx.x * blockDim.x + threadIdx.x;
  if (e < NE) atomicAdd(&deg[dst[e]], 1.0f);
}

__global__ void k_dinv(float* __restrict__ deg) {
  int i = blockIdx.x * blockDim.x + threadIdx.x;
  if (i < NN) { float d = deg[i]; deg[i] = (d > 0.0f) ? rsqrtf(d) : 0.0f; }
}

__global__ void k_norm(const int* __restrict__ src, const int* __restrict__ dst,
                       const float* __restrict__ dinv, float* __restrict__ nrm) {
  int e = blockIdx.x * blockDim.x + threadIdx.x;
  if (e < NE) nrm[e] = dinv[src[e]] * dinv[dst[e]];
}

// ---------------- fp32 WMMA GEMM: D[MxN] = A[MxK] * B[KxN] ----------------
// LDS-staged, compile-time KP (K padded up to a multiple of 4) and N.
// One block = N/16 waves; wave w computes the 16x16 tile at cols [16w, 16w+16).
// M must be a multiple of 16 (100000 = 6250 * 16).
//
// V_WMMA_F32_16X16X4_F32 fragments (ISA 7.12.2):
//   A (16x4):  lanes 0-15 rows, v[0]=K+0, v[1]=K+1; lanes 16-31: K+2, K+3.
//   B (4x16):  lanes 0-15 cols, v[0]=K+0, v[1]=K+1; lanes 16-31: K+2, K+3.
//   C/D:       v[i]: lanes 0-15 -> M=i, lanes 16-31 -> M=8+i.
template <int KP, int N>
__global__ void k_wmma_gemm(const float* __restrict__ A, int K,
                            const float* __restrict__ B,
                            float* __restrict__ D) {
  __shared__ float sA[16 * KP];   // row-major, zero-padded for k >= K
  __shared__ float sB[KP * N];    // row-major, zero-padded for k >= K

  const int tid  = threadIdx.x;
  const int nthr = 32 * (N / 16);
  const int row0 = blockIdx.x << 4;

  for (int idx = tid; idx < 16 * KP; idx += nthr) {
    const int r = idx / KP, k = idx - r * KP;
    sA[idx] = (k < K) ? A[(size_t)(row0 + r) * K + k] : 0.0f;
  }
  for (int idx = tid; idx < KP * N; idx += nthr) {
    const int k = idx / N, n = idx - k * N;
    sB[idx] = (k < K) ? B[(size_t)k * N + n] : 0.0f;
  }
  __syncthreads();

  const int lane = tid & 31;
  const int hl   = lane >> 4;         // lane half selects K+0/1 vs K+2/3
  const int l16  = lane & 15;
  const int col0 = (tid >> 5) << 4;   // wave id * 16

  v8f acc = {};
#pragma unroll
  for (int k = 0; k < KP; k += 4) {
    const int ka = k + 2 * hl;
    v2f a, b;
    a.x = sA[l16 * KP + ka];
    a.y = sA[l16 * KP + ka + 1];
    b.x = sB[ka * N + col0 + l16];
    b.y = sB[(ka + 1) * N + col0 + l16];
    acc = __builtin_amdgcn_wmma_f32_16x16x4_f32(
        false, a, false, b, (short)0, acc, false, false);
  }
#pragma unroll
  for (int i = 0; i < 8; ++i) {
    const int r = row0 + i + 8 * hl;
    D[(size_t)r * N + col0 + l16] = acc[i];
  }
}

// ---------------- aggregation ----------------

// agg[v,f] = dinv[v]^2 * h[v,f]   (self-loop contribution)
__global__ void k_selfloop_init(const float* __restrict__ h, const float* __restrict__ dinv,
                                float* __restrict__ agg, int lgF) {
  long long i = (long long)blockIdx.x * blockDim.x + threadIdx.x;
  long long tot = (long long)NN << lgF;
  if (i < tot) {
    int v = (int)(i >> lgF);
    float dv = dinv[v];
    agg[i] = dv * dv * h[i];
  }
}

// agg[dst, f..f+3] += norm[e] * h[src, f..f+3]
__global__ void k_edge_scatter(const float* __restrict__ h, const float* __restrict__ nrm,
                               const int* __restrict__ src, const int* __restrict__ dst,
                               float* __restrict__ agg, int F, int lgChunks) {
  long long t = (long long)blockIdx.x * blockDim.x + threadIdx.x;
  long long total = (long long)NE << lgChunks;
  if (t >= total) return;
  int e = (int)(t >> lgChunks);
  int f = ((int)t & ((1 << lgChunks) - 1)) << 2;
  int s = src[e], d = dst[e];
  float w = nrm[e];
  const float4 m = *reinterpret_cast<const float4*>(h + (size_t)s * F + f);
  float* o = agg + (size_t)d * F + f;
  atomicAdd(o + 0, w * m.x);
  atomicAdd(o + 1, w * m.y);
  atomicAdd(o + 2, w * m.z);
  atomicAdd(o + 3, w * m.w);
}

__global__ void k_bias_relu(float* __restrict__ agg, const float* __restrict__ b,
                            int F, int lgF) {
  long long i = (long long)blockIdx.x * blockDim.x + threadIdx.x;
  long long tot = (long long)NN << lgF;
  if (i < tot) {
    int f = (int)i & (F - 1);
    agg[i] = fmaxf(agg[i] + b[f], 0.0f);
  }
}

// ---------------- pooling + head ----------------

__global__ void k_zero(float* __restrict__ p, int n) {
  int i = blockIdx.x * blockDim.x + threadIdx.x;
  if (i < n) p[i] = 0.0f;
}

__global__ void k_pool(const float* __restrict__ h, const int* __restrict__ batch,
                       float* __restrict__ g) {
  long long i = (long long)blockIdx.x * blockDim.x + threadIdx.x;
  long long tot = (long long)NN * 32;
  if (i < tot) {
    int v = (int)(i >> 5);
    int f = (int)i & 31;
    atomicAdd(&g[(size_t)batch[v] * 32 + f], h[i]);
  }
}

__global__ void k_mlp(const float* __restrict__ g,
                      const float* __restrict__ A1, const float* __restrict__ c1,
                      const float* __restrict__ A2, const float* __restrict__ c2,
                      const float* __restrict__ A3, const float* __restrict__ c3,
                      const float* __restrict__ A4, const float* __restrict__ c4,
                      float* __restrict__ out) {
  __shared__ float sA1[32 * 32], sA2[32 * 16], sA3[16 * 8], sA4[8];
  __shared__ float sc1[32], sc2[16], sc3[8];
  int t = threadIdx.x;
  for (int i = t; i < 1024; i += 128) sA1[i] = A1[i];
  for (int i = t; i < 512;  i += 128) sA2[i] = A2[i];
  if (t < 128) sA3[t] = A3[t];
  if (t < 8)   sA4[t] = A4[t];
  if (t < 32)  sc1[t] = c1[t];
  if (t < 16)  sc2[t] = c2[t];
  if (t < 8)   sc3[t] = c3[t];
  __syncthreads();

  float vin[32], v1[32], v2[16], v3[8];
#pragma unroll
  for (int i = 0; i < 32; ++i) vin[i] = g[(size_t)t * 32 + i];
  for (int j = 0; j < 32; ++j) {
    float s = sc1[j];
#pragma unroll
    for (int i = 0; i < 32; ++i) s = fmaf(vin[i], sA1[i * 32 + j], s);
    v1[j] = fmaxf(s, 0.0f);
  }
  for (int j = 0; j < 16; ++j) {
    float s = sc2[j];
#pragma unroll
    for (int i = 0; i < 32; ++i) s = fmaf(v1[i], sA2[i * 16 + j], s);
    v2[j] = fmaxf(s, 0.0f);
  }
  for (int j = 0; j < 8; ++j) {
    float s = sc3[j];
#pragma unroll
    for (int i = 0; i < 16; ++i) s = fmaf(v2[i], sA3[i * 8 + j], s);
    v3[j] = fmaxf(s, 0.0f);
  }
  float s = c4[0];
#pragma unroll
  for (int i = 0; i < 8; ++i) s = fmaf(v3[i], sA4[i], s);
  out[t] = s;
}

// ---------------- launch ----------------

extern "C" void kernel_launch(void* const* d_in, const int* in_sizes, int n_in,
                              void* d_out, int out_size, void* d_ws, size_t ws_size,
                              hipStream_t stream) {
  const float* x     = (const float*)d_in[0];
  const int*   ei    = (const int*)d_in[1];
  const int*   src   = ei;         // edge_index[0]
  const int*   dst   = ei + NE;    // edge_index[1]
  const int*   batch = (const int*)d_in[2];
  const float* W1 = (const float*)d_in[3];  const float* b1 = (const float*)d_in[4];
  const float* W2 = (const float*)d_in[5];  const float* b2 = (const float*)d_in[6];
  const float* A1 = (const float*)d_in[7];  const float* c1 = (const float*)d_in[8];
  const float* A2 = (const float*)d_in[9];  const float* c2 = (const float*)d_in[10];
  const float* A3 = (const float*)d_in[11]; const float* c3 = (const float*)d_in[12];
  const float* A4 = (const float*)d_in[13]; const float* c4 = (const float*)d_in[14];
  float* out = (float*)d_out;

  float* ws  = (float*)d_ws;
  float* deg = ws; ws += NN;                 // becomes dinv in place
  float* nrm = ws; ws += NE;
  float* h1  = ws; ws += (size_t)NN * 64;
  float* ag1 = ws; ws += (size_t)NN * 64;    // becomes relu(agg + b1) in place
  float* h2  = ws; ws += (size_t)NN * 32;
  float* ag2 = ws; ws += (size_t)NN * 32;    // becomes relu(agg + b2) in place
  float* g   = ws; ws += NG * 32;

  const int T = 256;
  // degree + symmetric norm
  k_deg_init <<<(NN + T - 1) / T, T, 0, stream>>>(deg);
  k_deg_edges<<<(NE + T - 1) / T, T, 0, stream>>>(dst, deg);
  k_dinv     <<<(NN + T - 1) / T, T, 0, stream>>>(deg);
  k_norm     <<<(NE + T - 1) / T, T, 0, stream>>>(src, dst, deg, nrm);

  // layer 1: h1 = x @ W1 (K=47 padded to 48, N=64, 4 waves/block)
  k_wmma_gemm<48, 64><<<NN / 16, 128, 0, stream>>>(x, 47, W1, h1);
  k_selfloop_init<<<(int)(((long long)NN * 64 + T - 1) / T), T, 0, stream>>>(h1, deg, ag1, 6);
  k_edge_scatter <<<(int)(((long long)NE * 16 + T - 1) / T), T, 0, stream>>>(h1, nrm, src, dst, ag1, 64, 4);
  k_bias_relu    <<<(int)(((long long)NN * 64 + T - 1) / T), T, 0, stream>>>(ag1, b1, 64, 6);

  // layer 2: h2 = ag1 @ W2 (K=64, N=32, 2 waves/block)
  k_wmma_gemm<64, 32><<<NN / 16, 64, 0, stream>>>(ag1, 64, W2, h2);
  k_selfloop_init<<<(int)(((long long)NN * 32 + T - 1) / T), T, 0, stream>>>(h2, deg, ag2, 5);
  k_edge_scatter <<<(int)(((long long)NE * 8 + T - 1) / T), T, 0, stream>>>(h2, nrm, src, dst, ag2, 32, 3);
  k_bias_relu    <<<(int)(((long long)NN * 32 + T - 1) / T), T, 0, stream>>>(ag2, b2, 32, 5);

  // pool + MLP head
  k_zero<<<(NG * 32 + T - 1) / T, T, 0, stream>>>(g, NG * 32);
  k_pool<<<(int)(((long long)NN * 32 + T - 1) / T), T, 0, stream>>>(ag2, batch, g);
  k_mlp <<<1, 128, 0, stream>>>(g, A1, c1, A2, c2, A3, c3, A4, c4, out);
}